// PerceptionKernel_54674933678725
// MI455X (gfx1250) — compile-verified
//
#include <hip/hip_runtime.h>

// Differentiable logic-gate network forward (difflogic), MI455X / gfx1250.
// Memory-bound: 40 B/cell, ~42 MB total -> ~1.8 us floor at 23.3 TB/s.
// Layer1/2 matmuls on V_WMMA_F32_16X16X4_F32 with weights as the A operand,
// p/q neuron rows interleaved so each lane gets p_n,q_n in adjacent D regs:
// no shuffles, no predicated stores, no exec divergence in the hot loop.

typedef __attribute__((ext_vector_type(2))) float v2f;
typedef __attribute__((ext_vector_type(8))) float v8f;

#define WAVES_PER_BLOCK 8

// Gate g expressed in the {1, p, q, p*q} basis (difflogic ordering).
__device__ const float GT[16][4] = {
    {0.f, 0.f, 0.f, 0.f},   // FALSE
    {0.f, 0.f, 0.f, 1.f},   // AND
    {0.f, 1.f, 0.f,-1.f},   // A AND NOT B
    {0.f, 1.f, 0.f, 0.f},   // A
    {0.f, 0.f, 1.f,-1.f},   // NOT A AND B
    {0.f, 0.f, 1.f, 0.f},   // B
    {0.f, 1.f, 1.f,-2.f},   // XOR
    {0.f, 1.f, 1.f,-1.f},   // OR
    {1.f,-1.f,-1.f, 1.f},   // NOR
    {1.f,-1.f,-1.f, 2.f},   // XNOR
    {1.f, 0.f,-1.f, 0.f},   // NOT B
    {1.f, 0.f,-1.f, 1.f},   // A OR NOT B
    {1.f,-1.f, 0.f, 0.f},   // NOT A
    {1.f,-1.f, 0.f, 1.f},   // NOT A OR B
    {1.f, 0.f, 0.f,-1.f},   // NAND
    {1.f, 0.f, 0.f, 0.f},   // TRUE
};

__device__ __forceinline__ void softmax_row(const float* __restrict__ s, int len,
                                            float* __restrict__ d) {
  float m = s[0];
  for (int i = 1; i < len; ++i) m = fmaxf(m, s[i]);
  float sum = 0.f;
  for (int i = 0; i < len; ++i) { float e = expf(s[i] - m); d[i] = e; sum += e; }
  float inv = 1.f / sum;
  for (int i = 0; i < len; ++i) d[i] *= inv;
}

__device__ __forceinline__ void gate_coefs(const float* __restrict__ logits,
                                           float* __restrict__ c) {
  float w[16];
  softmax_row(logits, 16, w);
  float c0 = 0.f, c1 = 0.f, c2 = 0.f, c3 = 0.f;
#pragma unroll
  for (int g = 0; g < 16; ++g) {
    c0 += w[g] * GT[g][0]; c1 += w[g] * GT[g][1];
    c2 += w[g] * GT[g][2]; c3 += w[g] * GT[g][3];
  }
  c[0] = c0; c[1] = c1; c[2] = c2; c[3] = c3;
}

__global__ void __launch_bounds__(256)
lgn_forward(const float* __restrict__ X,
            const float* __restrict__ sa1, const float* __restrict__ sb1, const float* __restrict__ g1,
            const float* __restrict__ sa2, const float* __restrict__ sb2, const float* __restrict__ g2,
            const float* __restrict__ sa3, const float* __restrict__ sb3, const float* __restrict__ g3,
            const float* __restrict__ sa4, const float* __restrict__ sb4, const float* __restrict__ g4,
            float* __restrict__ out, int nCells)
{
  // W1'[m][k]: row 2n = softmax(sa1[n]), row 2n+1 = softmax(sb1[n]); k=0..8, 9..11 zero-pad
  __shared__ __align__(16) float sW1[16 * 12];
  // W2'[m][k]: rows 0..7 interleaved a2/b2 (4 neurons), rows 8..15 zero; k=0..7
  __shared__ __align__(16) float sW2[16 * 8];
  __shared__ float sC1[8 * 4];
  __shared__ float sC2[4 * 4];
  // [0..7]=A3, [8..15]=B3, [16..23]=C3, [24..25]=A4, [26..27]=B4, [28..31]=C4
  __shared__ float sL34[32];
  // per-wave h1 staging: 16 cells x 8 neurons
  __shared__ __align__(16) float sH[WAVES_PER_BLOCK * 128];

  // ---------------- per-block prep: softmaxes + gate-basis projection -------
  {
    const int t = threadIdx.x;
    float tmp[16];
    if (t < 16) {                               // W1' row t (interleaved p/q)
      const float* src = (t & 1) ? (sb1 + (t >> 1) * 9) : (sa1 + (t >> 1) * 9);
      softmax_row(src, 9, tmp);
      for (int k = 0; k < 12; ++k) sW1[t * 12 + k] = (k < 9) ? tmp[k] : 0.f;
    } else if (t < 32) {                        // W2' row (t-16)
      const int m = t - 16;
      if (m < 8) {
        const float* src = (m & 1) ? (sb2 + (m >> 1) * 8) : (sa2 + (m >> 1) * 8);
        softmax_row(src, 8, tmp);
        for (int k = 0; k < 8; ++k) sW2[m * 8 + k] = tmp[k];
      } else {
        for (int k = 0; k < 8; ++k) sW2[m * 8 + k] = 0.f;
      }
    } else if (t < 40) {
      gate_coefs(g1 + (t - 32) * 16, sC1 + (t - 32) * 4);
    } else if (t < 44) {
      gate_coefs(g2 + (t - 40) * 16, sC2 + (t - 40) * 4);
    } else if (t < 46) {
      const int n = t - 44;
      softmax_row(sa3 + n * 4, 4, tmp);
      for (int k = 0; k < 4; ++k) sL34[n * 4 + k] = tmp[k];
      softmax_row(sb3 + n * 4, 4, tmp);
      for (int k = 0; k < 4; ++k) sL34[8 + n * 4 + k] = tmp[k];
      gate_coefs(g3 + n * 16, sL34 + 16 + n * 4);
    } else if (t == 46) {
      softmax_row(sa4, 2, tmp); sL34[24] = tmp[0]; sL34[25] = tmp[1];
      softmax_row(sb4, 2, tmp); sL34[26] = tmp[0]; sL34[27] = tmp[1];
      gate_coefs(g4, sL34 + 28);
    }
  }
  __syncthreads();

  // ---------------- hot path ------------------------------------------------
  const int lane = threadIdx.x & 31;
  const int wIn  = threadIdx.x >> 5;
  const int m16  = lane & 15;     // A-row (weights) / B-col & D-col (cell)
  const int hi   = lane >> 4;     // half-wave: K-slice select / neuron-group select
  float* h1 = sH + wIn * 128;

  // A-operand fragments (uniform weights): reg r -> M = m16, K = 4s + r + 2*hi
  float aw1[3][2], aw2[2][2];
#pragma unroll
  for (int s = 0; s < 3; ++s) {
    aw1[s][0] = sW1[m16 * 12 + 4 * s + 2 * hi + 0];
    aw1[s][1] = sW1[m16 * 12 + 4 * s + 2 * hi + 1];
  }
#pragma unroll
  for (int s = 0; s < 2; ++s) {
    aw2[s][0] = sW2[m16 * 8 + 4 * s + 2 * hi + 0];
    aw2[s][1] = sW2[m16 * 8 + 4 * s + 2 * hi + 1];
  }
  // D rows r+8*hi are interleaved neurons: this lane owns neurons 4*hi+i of cell m16
  float cc1[4][4], cc2[4][4];
#pragma unroll
  for (int i = 0; i < 4; ++i)
#pragma unroll
    for (int j = 0; j < 4; ++j) {
      cc1[i][j] = sC1[(4 * hi + i) * 4 + j];
      cc2[i][j] = sC2[i * 4 + j];
    }
  float L34[32];
#pragma unroll
  for (int i = 0; i < 32; ++i) L34[i] = sL34[i];

  const int nChunks = nCells >> 5;                      // 32 cells per wave-chunk
  const int wid = blockIdx.x * WAVES_PER_BLOCK + wIn;
  const int nW  = gridDim.x * WAVES_PER_BLOCK;

  for (int c = wid; c < nChunks; c += nW) {
    const int base = c << 5;
    float hA[4], hB[4];                                 // tile0 / tile1 h2 (hi==0 lanes)
#pragma unroll
    for (int t = 0; t < 2; ++t) {
      // ---- layer1: D(16x16) = W1'(16x12) @ Xtile^T(12x16), 3 WMMA k-steps ----
      const float* xr = X + (size_t)(base + t * 16 + m16) * 9;
      v8f acc = {};
#pragma unroll
      for (int s = 0; s < 2; ++s) {
        v2f a, b;
        a.x = aw1[s][0]; a.y = aw1[s][1];
        b.x = xr[4 * s + 2 * hi]; b.y = xr[4 * s + 2 * hi + 1];   // K in 0..7: valid
        acc = __builtin_amdgcn_wmma_f32_16x16x4_f32(false, a, false, b,
                                                    (short)0, acc, false, false);
      }
      {
        // K = 8..11: only k=8 has nonzero weights -> load x[8] unconditionally,
        // don't-care B values pair with zero-padded A rows. No predication.
        v2f a, b;
        a.x = aw1[2][0]; a.y = aw1[2][1];
        const float x8 = xr[8];
        b.x = x8; b.y = x8;
        acc = __builtin_amdgcn_wmma_f32_16x16x4_f32(false, a, false, b,
                                                    (short)0, acc, false, false);
      }
      // gate mix: p_n = acc[2i], q_n = acc[2i+1] in-lane; 4 unique h1 per lane
      float hv0, hv1, hv2, hv3;
      {
        const float p0 = acc[0], q0 = acc[1], p1 = acc[2], q1 = acc[3];
        const float p2 = acc[4], q2 = acc[5], p3 = acc[6], q3 = acc[7];
        hv0 = cc1[0][0] + cc1[0][1] * p0 + cc1[0][2] * q0 + cc1[0][3] * p0 * q0;
        hv1 = cc1[1][0] + cc1[1][1] * p1 + cc1[1][2] * q1 + cc1[1][3] * p1 * q1;
        hv2 = cc1[2][0] + cc1[2][1] * p2 + cc1[2][2] * q2 + cc1[2][3] * p2 * q2;
        hv3 = cc1[3][0] + cc1[3][1] * p3 + cc1[3][2] * q3 + cc1[3][3] * p3 * q3;
      }
      *(float4*)(h1 + m16 * 8 + 4 * hi) = make_float4(hv0, hv1, hv2, hv3);
      asm volatile("s_wait_dscnt 0x0" ::: "memory");    // h1 visible wave-wide

      // ---- layer2: D = W2'(16x8) @ h1^T(8x16), 2 WMMA k-steps ----
      v8f acc2 = {};
#pragma unroll
      for (int s = 0; s < 2; ++s) {
        v2f a, b;
        a.x = aw2[s][0]; a.y = aw2[s][1];
        const float2 hb = *(const float2*)(h1 + m16 * 8 + 4 * s + 2 * hi);
        b.x = hb.x; b.y = hb.y;
        acc2 = __builtin_amdgcn_wmma_f32_16x16x4_f32(false, a, false, b,
                                                     (short)0, acc2, false, false);
      }
      // h2 for cell m16 (meaningful in hi==0 lanes; hi==1 rows are zero-weight)
#pragma unroll
      for (int i = 0; i < 4; ++i) {
        const float pv = acc2[2 * i], qv = acc2[2 * i + 1];
        const float h = cc2[i][0] + cc2[i][1] * pv + cc2[i][2] * qv + cc2[i][3] * pv * qv;
        if (t == 0) hA[i] = h; else hB[i] = h;
      }
    }

    // pair tiles across wave halves: lane hi*16+m16 finalizes cell base+lane
    float f[4];
#pragma unroll
    for (int i = 0; i < 4; ++i) {
      const float up = __shfl_xor(hB[i], 16, 32);       // tile1 -> upper half
      f[i] = hi ? up : hA[i];
    }
    // ---- layers 3+4 in registers, fully coalesced 128B store per wave ----
    float o3[2];
#pragma unroll
    for (int u = 0; u < 2; ++u) {
      const float p = L34[u*4+0]*f[0] + L34[u*4+1]*f[1] + L34[u*4+2]*f[2] + L34[u*4+3]*f[3];
      const float q = L34[8+u*4+0]*f[0] + L34[8+u*4+1]*f[1] + L34[8+u*4+2]*f[2] + L34[8+u*4+3]*f[3];
      o3[u] = L34[16+u*4+0] + L34[16+u*4+1]*p + L34[16+u*4+2]*q + L34[16+u*4+3]*p*q;
    }
    const float p4 = L34[24]*o3[0] + L34[25]*o3[1];
    const float q4 = L34[26]*o3[0] + L34[27]*o3[1];
    out[base + lane] = L34[28] + L34[29]*p4 + L34[30]*q4 + L34[31]*p4*q4;
  }
}

extern "C" void kernel_launch(void* const* d_in, const int* in_sizes, int n_in,
                              void* d_out, int out_size, void* d_ws, size_t ws_size,
                              hipStream_t stream) {
  (void)n_in; (void)out_size; (void)d_ws; (void)ws_size;
  const float* X   = (const float*)d_in[0];
  const float* sa1 = (const float*)d_in[1];
  const float* sb1 = (const float*)d_in[2];
  const float* g1  = (const float*)d_in[3];
  const float* sa2 = (const float*)d_in[4];
  const float* sb2 = (const float*)d_in[5];
  const float* g2  = (const float*)d_in[6];
  const float* sa3 = (const float*)d_in[7];
  const float* sb3 = (const float*)d_in[8];
  const float* g3  = (const float*)d_in[9];
  const float* sa4 = (const float*)d_in[10];
  const float* sb4 = (const float*)d_in[11];
  const float* g4  = (const float*)d_in[12];
  float* out = (float*)d_out;

  const int nCells  = in_sizes[0] / 9;     // B = 1048576 (divisible by 32)
  const int nChunks = nCells >> 5;
  int grid = 1024;                          // 8192 waves -> ~4 chunks/wave
  const int maxGrid = (nChunks + WAVES_PER_BLOCK - 1) / WAVES_PER_BLOCK;
  if (grid > maxGrid) grid = maxGrid;
  if (grid < 1) grid = 1;

  lgn_forward<<<grid, 256, 0, stream>>>(X, sa1, sb1, g1, sa2, sb2, g2,
                                        sa3, sb3, g3, sa4, sb4, g4,
                                        out, nCells);
}